// RoIPool_35141422416215
// MI455X (gfx1250) — compile-verified
//
#include <hip/hip_runtime.h>
#include <hip/hip_bf16.h>
#include <stdint.h>
#include <math.h>

// RoI max pooling for MI455X (gfx1250).
// features: [B=2, C=256, H=30, W=40] f32; rois: [R, 5] f32; out: [R, C, 7, 7] f32.
// One block per RoI, 8 wave32 waves; each wave stages one channel's RoI window
// into LDS with the Tensor Data Mover (double buffered, TENSORcnt pipelined),
// then computes the 49 bin maxima from LDS.

#define POOLN 7
#define NBINS (POOLN * POOLN)
#define CC 256
#define HH 30
#define WW 40
#define ROI_SCALE 0.0625f
#define MAXWIN (HH * WW)   // 1200 floats: max clipped window per channel
#define NWAVES 8

typedef unsigned int v4u __attribute__((ext_vector_type(4)));
typedef int          v8i __attribute__((ext_vector_type(8)));
typedef int          v4i __attribute__((ext_vector_type(4)));

#if defined(__has_builtin)
#if __has_builtin(__builtin_amdgcn_tensor_load_to_lds) && \
    __has_builtin(__builtin_amdgcn_s_wait_tensorcnt)
#define USE_TDM 1
#endif
#endif

#ifdef USE_TDM
// Issue one TDM 2-D tile load: global (winW x winH tile, row stride WW elems,
// 4B elements) -> packed LDS at lds_byte_addr. Tracked by TENSORcnt.
// This toolchain exposes the 6-arg builtin:
//   (uint32x4 g0, int32x8 g1, int32x4 g2, int32x4 g3, int32x8 extra, i32 cpol)
__device__ __forceinline__ void tdm_load_tile(uint32_t lds_byte_addr,
                                              uint64_t gaddr,
                                              int winW, int winH) {
  // D# group 0 (128b): count=1 | lds_addr | global_addr[56:0] | type=2
  v4u g0;
  g0.x = 1u;                                   // count=1 valid descriptor
  g0.y = lds_byte_addr;                        // LDS byte address
  g0.z = (uint32_t)gaddr;                      // global_addr[31:0]
  g0.w = ((uint32_t)(gaddr >> 32) & 0x01FFFFFFu) | (2u << 30); // addr[56:32], type=2

  // D# group 1 (256b), window treated as the tensor (no OOB):
  //   data_size=2 (4B), tensor_dim0=winW, tensor_dim1=winH,
  //   tile_dim0=winW, tile_dim1=winH, tile_dim2=0,
  //   tensor_dim0_stride=WW, tensor_dim1_stride=HH*WW (unused for 2-D tile).
  unsigned uW = (unsigned)winW, uH = (unsigned)winH;
  v8i g1;
  g1[0] = (int)(2u << 16);                                   // data_size=4B
  g1[1] = (int)((uW & 0xFFFFu) << 16);                       // tensor_dim0 lo16
  g1[2] = (int)(((uW >> 16) & 0xFFFFu) | ((uH & 0xFFFFu) << 16)); // dim0 hi | dim1 lo
  g1[3] = (int)(((uH >> 16) & 0xFFFFu) | ((uW & 0xFFFFu) << 16)); // dim1 hi | tile_dim0
  g1[4] = (int)(uH & 0xFFFFu);                               // tile_dim1, tile_dim2=0
  g1[5] = (int)WW;                                           // dim0_stride lo32
  g1[6] = (int)(((unsigned)(HH * WW) & 0xFFFFu) << 16);      // dim0_stride hi=0 | dim1_stride lo16
  g1[7] = 0;                                                 // dim1_stride hi
  v4i z4 = {0, 0, 0, 0};                                     // groups 2/3: 2-D tile
  v8i z8 = {0, 0, 0, 0, 0, 0, 0, 0};
  __builtin_amdgcn_tensor_load_to_lds(g0, g1, z4, z4, z8, 0);
}
#endif

__global__ __launch_bounds__(NWAVES * 32) void
roipool_tdm_kernel(const float* __restrict__ feat,
                   const float* __restrict__ rois,
                   float* __restrict__ out) {
  __shared__ float lds[NWAVES * 2 * MAXWIN];  // 76.8 KB, double buffer per wave

  const int r    = blockIdx.x;
  const int lane = threadIdx.x & 31;
  const int wave = threadIdx.x >> 5;

  // ---- RoI geometry (block-uniform), replicating the reference math ----
  const float* roi = rois + (size_t)r * 5;
  const int   bidx = (int)roi[0];
  const float fx1  = rintf(roi[1] * ROI_SCALE);   // rintf == round-half-even
  const float fy1  = rintf(roi[2] * ROI_SCALE);
  const float fx2  = rintf(roi[3] * ROI_SCALE);
  const float fy2  = rintf(roi[4] * ROI_SCALE);
  const float roiw = fmaxf(fx2 - fx1 + 1.0f, 1.0f);
  const float roih = fmaxf(fy2 - fy1 + 1.0f, 1.0f);
  const float binw = roiw * (1.0f / POOLN);
  const float binh = roih * (1.0f / POOLN);

  const int iy1 = (int)fy1, ix1 = (int)fx1;
  const int rih = (int)roih, riw = (int)roiw;
  // Conservative union window of all bins, clipped to the feature map.
  const int h0 = min(max(iy1, 0), HH);
  const int h1 = min(max(iy1 + rih + 1, 0), HH);
  const int w0 = min(max(ix1, 0), WW);
  const int w1 = min(max(ix1 + riw + 1, 0), WW);
  const int winH = h1 - h0;
  const int winW = w1 - w0;

  // ---- Per-lane bin boundaries (bins b = lane and lane+32) ----
  int  bs_h[2], be_h[2], bs_w[2], be_w[2];
  bool bok[2] = {false, false};
#pragma unroll
  for (int k = 0; k < 2; ++k) {
    const int b = lane + 32 * k;
    if (b < NBINS) {
      const int ph = b / POOLN, pw = b % POOLN;
      bs_h[k] = (int)fminf(fmaxf(floorf((float)ph * binh) + fy1, 0.0f), (float)HH);
      be_h[k] = (int)fminf(fmaxf(ceilf((float)(ph + 1) * binh) + fy1, 0.0f), (float)HH);
      bs_w[k] = (int)fminf(fmaxf(floorf((float)pw * binw) + fx1, 0.0f), (float)WW);
      be_w[k] = (int)fminf(fmaxf(ceilf((float)(pw + 1) * binw) + fx1, 0.0f), (float)WW);
      bok[k] = true;
    }
  }

  // Fully-clipped-away RoI: every bin is empty -> zeros.
  if (winW <= 0 || winH <= 0) {
    for (int c = wave; c < CC; c += NWAVES) {
      float* oc = out + ((size_t)r * CC + c) * NBINS;
#pragma unroll
      for (int k = 0; k < 2; ++k)
        if (bok[k]) oc[lane + 32 * k] = 0.0f;
    }
    return;
  }

  const float* fbase = feat + (size_t)bidx * CC * HH * WW;

#ifdef USE_TDM
  float* slot0 = lds + (size_t)wave * 2 * MAXWIN;
  float* slot1 = slot0 + MAXWIN;
  const uint32_t ldsOff0 = (uint32_t)(uintptr_t)slot0;  // flat low 32 = LDS byte offset
  const uint32_t ldsOff1 = (uint32_t)(uintptr_t)slot1;
  const uint64_t gbase   = (uint64_t)(uintptr_t)fbase;
  const uint64_t winOff  = ((uint64_t)h0 * WW + (uint64_t)w0) * 4u;

  // Preload channel `wave` into buffer 0.
  tdm_load_tile(ldsOff0, gbase + (uint64_t)wave * (HH * WW * 4u) + winOff,
                winW, winH);

  const int iters = CC / NWAVES;  // 32
  for (int i = 0; i < iters; ++i) {
    const int    c   = wave + i * NWAVES;
    const float* cur = (i & 1) ? slot1 : slot0;

    // Current tile landed.
    __builtin_amdgcn_s_wait_tensorcnt(0);

    // Overlap: kick off next channel's tile into the other buffer. Make sure
    // no in-flight LDS reads of that buffer remain before TDM rewrites it.
    if (i + 1 < iters) {
      asm volatile("s_wait_dscnt 0" ::: "memory");
      const uint32_t nlds = ((i + 1) & 1) ? ldsOff1 : ldsOff0;
      tdm_load_tile(nlds,
                    gbase + (uint64_t)(c + NWAVES) * (HH * WW * 4u) + winOff,
                    winW, winH);
    }

    float* oc = out + ((size_t)r * CC + c) * NBINS;
#pragma unroll
    for (int k = 0; k < 2; ++k) {
      if (!bok[k]) continue;
      const int hs = bs_h[k], he = be_h[k], ws = bs_w[k], we = be_w[k];
      float m = -__builtin_inff();
      for (int h = hs; h < he; ++h) {
        const float* row = cur + (h - h0) * winW;
        for (int w = ws; w < we; ++w) m = fmaxf(m, row[w - w0]);
      }
      oc[lane + 32 * k] = (hs < he && ws < we) ? m : 0.0f;
    }
  }
#else
  // Fallback: direct global reads (features stay resident in 192MB L2 anyway).
  for (int c = wave; c < CC; c += NWAVES) {
    const float* fc = fbase + (size_t)c * HH * WW;
    float* oc = out + ((size_t)r * CC + c) * NBINS;
#pragma unroll
    for (int k = 0; k < 2; ++k) {
      if (!bok[k]) continue;
      const int hs = bs_h[k], he = be_h[k], ws = bs_w[k], we = be_w[k];
      float m = -__builtin_inff();
      for (int h = hs; h < he; ++h)
        for (int w = ws; w < we; ++w) m = fmaxf(m, fc[h * WW + w]);
      oc[lane + 32 * k] = (hs < he && ws < we) ? m : 0.0f;
    }
  }
#endif
}

extern "C" void kernel_launch(void* const* d_in, const int* in_sizes, int n_in,
                              void* d_out, int out_size, void* d_ws, size_t ws_size,
                              hipStream_t stream) {
  (void)n_in; (void)out_size; (void)d_ws; (void)ws_size;
  const float* feat = (const float*)d_in[0];
  const float* rois = (const float*)d_in[1];
  float* out = (float*)d_out;
  const int R = in_sizes[1] / 5;  // rois: [R, 5]
  roipool_tdm_kernel<<<R, NWAVES * 32, 0, stream>>>(feat, rois, out);
}